// SpecAugment_37735582662926
// MI455X (gfx1250) — compile-verified
//
#include <hip/hip_runtime.h>

// SpecAugment: out = (x + noise*0.04) * keep(b,f,t)
// B=128, C=3, F=128, T=1000; NF=NT=2 masks per sample.
// Memory-bound: ~590 MB moved -> ~25 us roofline at 23.3 TB/s.

#define NOISE_STD 0.04f

typedef float v4f __attribute__((ext_vector_type(4)));  // native clang vector

constexpr int Bc = 128;
constexpr int Cc = 3;
constexpr int Fc = 128;   // power of two
constexpr int Tc = 1000;  // divisible by 4
constexpr int NFm = 2;
constexpr int NTm = 2;
constexpr int ROWS = Bc * Cc * Fc;  // 49152 rows of length T
constexpr int TVEC = Tc / 4;        // 250 float4 per row

__global__ __launch_bounds__(256) void specaug_kernel(
    const float* __restrict__ x, const float* __restrict__ noise,
    const int* __restrict__ f0, const int* __restrict__ f_len,
    const int* __restrict__ t0, const int* __restrict__ t_len,
    float* __restrict__ out)
{
    const int row = blockIdx.x;           // (b, c, f) flattened; uniform
    const int f   = row & (Fc - 1);       // F = 128
    const int b   = row / (Cc * Fc);      // uniform -> SALU mul-shift

    // Frequency mask: uniform across the whole block (depends on b, f only).
    bool fmask = false;
#pragma unroll
    for (int i = 0; i < NFm; ++i) {
        const int s = f0[b * NFm + i];
        const int l = f_len[b * NFm + i];
        fmask |= ((unsigned)(f - s) < (unsigned)l);
    }

    // Time-mask parameters: uniform across the block; per-lane test below.
    const int s0 = t0[b * NTm + 0];
    const int l0 = t_len[b * NTm + 0];
    const int s1 = t0[b * NTm + 1];
    const int l1 = t_len[b * NTm + 1];

    const int tid = threadIdx.x;
    if (tid >= TVEC) return;              // 250 of 256 lanes active

    const int elem   = row * Tc + tid * 4;   // < 49.2M, fits in int
    const int t_base = tid * 4;

    // Streaming data: read-once / write-once -> non-temporal (CDNA5 TH=NT)
    const v4f xv = __builtin_nontemporal_load((const v4f*)(x + elem));
    const v4f nv = __builtin_nontemporal_load((const v4f*)(noise + elem));

    v4f r;
#define SPECAUG_ELEM(comp, j)                                                  \
    {                                                                          \
        const int t = t_base + (j);                                            \
        const bool tm = ((unsigned)(t - s0) < (unsigned)l0) |                  \
                        ((unsigned)(t - s1) < (unsigned)l1);                   \
        const float k = (fmask | tm) ? 0.0f : 1.0f;                            \
        r.comp = (xv.comp + nv.comp * NOISE_STD) * k;                          \
    }
    SPECAUG_ELEM(x, 0)
    SPECAUG_ELEM(y, 1)
    SPECAUG_ELEM(z, 2)
    SPECAUG_ELEM(w, 3)
#undef SPECAUG_ELEM

    __builtin_nontemporal_store(r, (v4f*)(out + elem));
}

extern "C" void kernel_launch(void* const* d_in, const int* in_sizes, int n_in,
                              void* d_out, int out_size, void* d_ws, size_t ws_size,
                              hipStream_t stream) {
    (void)in_sizes; (void)n_in; (void)out_size; (void)d_ws; (void)ws_size;
    const float* x     = (const float*)d_in[0];
    const float* noise = (const float*)d_in[1];
    const int*   f0    = (const int*)d_in[2];
    const int*   f_len = (const int*)d_in[3];
    const int*   t0    = (const int*)d_in[4];
    const int*   t_len = (const int*)d_in[5];
    float*       out   = (float*)d_out;

    dim3 grid(ROWS);   // 49152 blocks, one (b,c,f) row each
    dim3 block(256);   // 8 wave32 waves
    specaug_kernel<<<grid, block, 0, stream>>>(x, noise, f0, f_len, t0, t_len, out);
}